// MultiplicativeCrossAttention_28845000360515
// MI455X (gfx1250) — compile-verified
//
#include <hip/hip_runtime.h>
#include <hip/hip_bf16.h>

// Problem sizes (fixed by the reference)
#define BATCH   32
#define SEQ     4096
#define QDIM    512
#define KVDIM   512

// Fused-pass tiling
#define NCHUNK   16                    // S-chunks per batch
#define ROWS_PER_CHUNK (SEQ / NCHUNK)  // 256
#define WAVES    8
#define TPB      (WAVES * 32)          // 256 threads = 8 wave32

// Workspace layout (floats)
#define WQ_OFF      0                        // [BATCH][KVDIM]
#define PART_STRIDE (KVDIM + 8)              // acc[512] + m + l (+pad)
#define PART_OFF    (BATCH * KVDIM)          // [BATCH*NCHUNK][PART_STRIDE]

typedef float v2f __attribute__((ext_vector_type(2)));
typedef float v8f __attribute__((ext_vector_type(8)));

// ---------------------------------------------------------------------------
// Kernel A: wq[b,k] = sum_q query[b,q] * weight[q,k]   (32x512x512 fp32 GEMM)
// One wave per 16x16 output tile, V_WMMA_F32_16X16X4_F32, K stepped by 4.
// A-frag (16x4 f32, ISA 7.12.2): lane<16 holds {K=0,K=1} of row M=lane,
// lane>=16 holds {K=2,K=3} of row M=lane-16. B-frag symmetric (col = lane%16).
// C/D: VGPR r -> M=r (lanes 0-15) / M=r+8 (lanes 16-31), N = lane%16.
// ---------------------------------------------------------------------------
__global__ void __launch_bounds__(32)
wq_gemm_wmma(const float* __restrict__ query,
             const float* __restrict__ weight,
             float* __restrict__ wq) {
  const int tile = blockIdx.x;      // 64 tiles: 2 (M) x 32 (N)
  const int tM   = tile >> 5;
  const int tN   = tile & 31;
  const int lane = threadIdx.x;
  const int half = lane >> 4;       // 0 or 1
  const int lr   = lane & 15;

  const int aRow = tM * 16 + lr;    // query row (batch index)
  const int bCol = tN * 16 + lr;    // weight/output column (k index)

  v8f c = {};
  for (int k0 = 0; k0 < QDIM; k0 += 4) {
    const int ka = k0 + half * 2;
    v2f a, b;
    a[0] = query[aRow * QDIM + ka];
    a[1] = query[aRow * QDIM + ka + 1];
    b[0] = weight[(size_t)ka * KVDIM + bCol];
    b[1] = weight[(size_t)(ka + 1) * KVDIM + bCol];
    // 8 args: (neg_a, A, neg_b, B, c_mod, C, reuse_a, reuse_b)
    c = __builtin_amdgcn_wmma_f32_16x16x4_f32(false, a, false, b,
                                              (short)0, c, false, false);
  }
#pragma unroll
  for (int r = 0; r < 8; ++r) {
    wq[(tM * 16 + half * 8 + r) * KVDIM + tN * 16 + lr] = c[r];
  }
}

// ---------------------------------------------------------------------------
// Kernel B: fused masked-score + online-softmax + weighted accumulate.
// One block = (batch b, S-chunk). Each wave streams whole rows of keyvalues
// (read exactly once from HBM), keeps the row in registers, maintains a
// flash-style (m, l, acc[512]) partial. Wave partials combined in LDS; one
// chunk partial written to workspace.
// ---------------------------------------------------------------------------
__global__ void __launch_bounds__(TPB)
attn_partial(const float* __restrict__ kv,
             const int*   __restrict__ mask,
             const float* __restrict__ wq,
             float*       __restrict__ part) {
  __shared__ float s_wq[KVDIM];
  __shared__ float s_acc[WAVES][KVDIM];
  __shared__ float s_m[WAVES];
  __shared__ float s_l[WAVES];

  const int b     = blockIdx.x / NCHUNK;
  const int chunk = blockIdx.x % NCHUNK;
  const int tid   = threadIdx.x;
  const int wave  = tid >> 5;
  const int lane  = tid & 31;
  const float NEG_INF = -__builtin_inff();

  for (int i = tid; i < KVDIM; i += TPB) s_wq[i] = wq[b * KVDIM + i];
  __syncthreads();

  // This lane's 16 wq elements (layout matches the kv row loads below).
  float4 w4[4];
#pragma unroll
  for (int q = 0; q < 4; ++q)
    w4[q] = *(const float4*)&s_wq[q * 128 + lane * 4];

  float  m = NEG_INF, l = 0.f;
  float4 acc[4] = {};

  const int sBase = chunk * ROWS_PER_CHUNK;
  const int iters = ROWS_PER_CHUNK / WAVES;   // 32
  for (int it = 0; it < iters; ++it) {
    const int s = sBase + it * WAVES + wave;
    const float* __restrict__ row = kv + ((size_t)b * SEQ + s) * KVDIM;
    if (it + 1 < iters)                        // speculative stream prefetch
      __builtin_prefetch(row + WAVES * KVDIM, 0, 0);

    float4 r4[4];
#pragma unroll
    for (int q = 0; q < 4; ++q)                // 4x b128, fully coalesced
      r4[q] = *(const float4*)&row[q * 128 + lane * 4];

    float d = 0.f;
#pragma unroll
    for (int q = 0; q < 4; ++q) {
      d = fmaf(r4[q].x, w4[q].x, d);
      d = fmaf(r4[q].y, w4[q].y, d);
      d = fmaf(r4[q].z, w4[q].z, d);
      d = fmaf(r4[q].w, w4[q].w, d);
    }
    // wave32 butterfly reduction -> every lane holds the full dot product
#pragma unroll
    for (int off = 16; off > 0; off >>= 1) d += __shfl_xor(d, off, 32);

    if (mask[b * SEQ + s] != 0) {
      const float mN    = fmaxf(m, d);
      const float scale = __expf(m - mN);      // m==-inf -> 0
      const float p     = __expf(d - mN);
      l = l * scale + p;
#pragma unroll
      for (int q = 0; q < 4; ++q) {
        acc[q].x = fmaf(p, r4[q].x, acc[q].x * scale);
        acc[q].y = fmaf(p, r4[q].y, acc[q].y * scale);
        acc[q].z = fmaf(p, r4[q].z, acc[q].z * scale);
        acc[q].w = fmaf(p, r4[q].w, acc[q].w * scale);
      }
      m = mN;
    }
  }

  // publish wave partials
#pragma unroll
  for (int q = 0; q < 4; ++q)
    *(float4*)&s_acc[wave][q * 128 + lane * 4] = acc[q];
  if (lane == 0) { s_m[wave] = m; s_l[wave] = l; }
  __syncthreads();

  // combine 8 waves -> one chunk partial
  float mB = NEG_INF;
#pragma unroll
  for (int w = 0; w < WAVES; ++w) mB = fmaxf(mB, s_m[w]);
  float wgt[WAVES];
  float lB = 0.f;
#pragma unroll
  for (int w = 0; w < WAVES; ++w) {
    const float wi = (s_m[w] == NEG_INF) ? 0.f : __expf(s_m[w] - mB);
    wgt[w] = wi;
    lB += wi * s_l[w];
  }
  float* P = part + (size_t)(b * NCHUNK + chunk) * PART_STRIDE;
  for (int e = tid; e < KVDIM; e += TPB) {
    float v = 0.f;
#pragma unroll
    for (int w = 0; w < WAVES; ++w) v = fmaf(wgt[w], s_acc[w][e], v);
    P[e] = v;
  }
  if (tid == 0) { P[KVDIM] = mB; P[KVDIM + 1] = lB; }
}

// ---------------------------------------------------------------------------
// Kernel C: combine the NCHUNK partials per batch, normalize, write a_t.
// ---------------------------------------------------------------------------
__global__ void __launch_bounds__(256)
attn_combine(const float* __restrict__ part, float* __restrict__ out) {
  const int b   = blockIdx.x;
  const int tid = threadIdx.x;
  const float NEG_INF = -__builtin_inff();

  float mG = NEG_INF;
#pragma unroll
  for (int c = 0; c < NCHUNK; ++c)
    mG = fmaxf(mG, part[(size_t)(b * NCHUNK + c) * PART_STRIDE + KVDIM]);

  float wc[NCHUNK];
  float lG = 0.f;
#pragma unroll
  for (int c = 0; c < NCHUNK; ++c) {
    const size_t base = (size_t)(b * NCHUNK + c) * PART_STRIDE;
    const float  mi   = part[base + KVDIM];
    const float  wi   = (mi == NEG_INF) ? 0.f : __expf(mi - mG);
    wc[c] = wi;
    lG += wi * part[base + KVDIM + 1];
  }
  const float inv = 1.f / lG;

  for (int e = tid; e < KVDIM; e += 256) {
    float v = 0.f;
#pragma unroll
    for (int c = 0; c < NCHUNK; ++c)
      v = fmaf(wc[c], part[(size_t)(b * NCHUNK + c) * PART_STRIDE + e], v);
    out[b * KVDIM + e] = v * inv;
  }
}

// ---------------------------------------------------------------------------
extern "C" void kernel_launch(void* const* d_in, const int* in_sizes, int n_in,
                              void* d_out, int out_size, void* d_ws, size_t ws_size,
                              hipStream_t stream) {
  const float* query  = (const float*)d_in[0];   // [32, 512]
  const float* kv     = (const float*)d_in[1];   // [32, 4096, 512]
  const int*   mask   = (const int*)  d_in[2];   // [32, 4096]
  const float* weight = (const float*)d_in[3];   // [512, 512]
  float* out = (float*)d_out;                    // [32, 512]

  float* ws   = (float*)d_ws;                    // ~1.13 MB used
  float* wq   = ws + WQ_OFF;
  float* part = ws + PART_OFF;

  // A: wq = query @ weight  (fp32 WMMA, 64 waves)
  wq_gemm_wmma<<<64, 32, 0, stream>>>(query, weight, wq);
  // B: single streaming pass over keyvalues with online softmax
  attn_partial<<<BATCH * NCHUNK, TPB, 0, stream>>>(kv, mask, wq, part);
  // C: reduce chunk partials -> a_t
  attn_combine<<<BATCH, 256, 0, stream>>>(part, out);
}